// PlainTTT_3204045603558
// MI455X (gfx1250) — compile-verified
//
#include <hip/hip_runtime.h>

typedef float v2f __attribute__((ext_vector_type(2)));
typedef float v8f __attribute__((ext_vector_type(8)));

#define HH     64
#define HH2    128
#define LL     4096
#define BBATCH 64
#define TSTEPS 2047          // len(range(0, 4096-3, 2))
#define ROWP   132           // padded LDS row stride (floats) to dodge bank conflicts
#define LRATE  0.05f

// D = A(16x4 f32) * B(4x16 f32) + C(16x16 f32), exact f32 on the matrix pipe.
__device__ __forceinline__ v8f wmma_f32_16x16x4(v2f a, v2f b, v8f c) {
  return __builtin_amdgcn_wmma_f32_16x16x4_f32(false, a, false, b, (short)0, c,
                                               false, false);
}

// ---------------------------------------------------------------------------
// Phase 1: per-token MLP + LayerNorm -> hs[B*L][H], one wave per 16 tokens.
// A-operand layout (16x4 f32): lane L, reg r -> A[L%16][4*kt + 2*(L/16) + r]
// B-operand layout (4x16 f32): lane L, reg r -> B[4*kt + 2*(L/16) + r][L%16]
// C/D layout (16x16 f32):      lane L, reg r -> C[8*(L/16) + r][L%16]
// ---------------------------------------------------------------------------
__global__ __launch_bounds__(128) void tok_hs_kernel(
    const int* __restrict__ seq, const float* __restrict__ embed,
    const float* __restrict__ ffw1, const float* __restrict__ ffb1,
    const float* __restrict__ ffw2, const float* __restrict__ ffb2,
    const float* __restrict__ lng, const float* __restrict__ lnb,
    float* __restrict__ hs) {
  __shared__ float stg_all[4][16 * ROWP];   // per-wave staging, 33.8 KB total
  const int wave  = threadIdx.x >> 5;
  const int lane  = threadIdx.x & 31;
  const int mrow  = lane & 15;              // A matrix row / C matrix col
  const int khalf = lane >> 4;              // which K/M half this lane serves
  float* st = stg_all[wave];

  const long tile = (long)blockIdx.x * 4 + wave;  // 16 tokens per tile
  const long tok  = tile * 16 + mrow;

  const int sid = seq[tok];
  const float* erow = embed + sid * HH;     // 16 KB table -> L0 resident

  // ---- layer 1: h1 = relu(e @ ffw1^T + b1)  (16 x 64) x (64 x 128) ----
  v2f a1[16];
#pragma unroll
  for (int kt = 0; kt < 16; ++kt)
    a1[kt] = *(const v2f*)(erow + 4 * kt + 2 * khalf);

#pragma unroll
  for (int nt = 0; nt < 8; ++nt) {
    const int ncol = nt * 16 + mrow;
    const float bv = ffb1[ncol];
    v8f acc = {bv, bv, bv, bv, bv, bv, bv, bv};
    const float* wrow = ffw1 + ncol * HH;   // B[k][n] = ffw1[n][k]
#pragma unroll
    for (int kt = 0; kt < 16; ++kt) {
      v2f bop = *(const v2f*)(wrow + 4 * kt + 2 * khalf);
      acc = wmma_f32_16x16x4(a1[kt], bop, acc);
    }
#pragma unroll
    for (int r = 0; r < 8; ++r)             // relu, stage h1[m][n] for transpose
      st[(8 * khalf + r) * ROWP + ncol] = fmaxf(acc[r], 0.0f);
  }

  // ---- layer 2: f = h1 @ ffw2^T + b2  (16 x 128) x (128 x 64) ----
  v2f a2[32];                               // LDS does the C->A relayout
#pragma unroll
  for (int kt = 0; kt < 32; ++kt)
    a2[kt] = *(const v2f*)(st + mrow * ROWP + 4 * kt + 2 * khalf);

  v8f f2[4];
#pragma unroll
  for (int nt = 0; nt < 4; ++nt) {
    const int ncol = nt * 16 + mrow;
    const float bv = ffb2[ncol];
    v8f acc = {bv, bv, bv, bv, bv, bv, bv, bv};
    const float* wrow = ffw2 + ncol * HH2;  // B[k][n] = ffw2[n][k]
#pragma unroll
    for (int kt = 0; kt < 32; ++kt) {
      v2f bop = *(const v2f*)(wrow + 4 * kt + 2 * khalf);
      acc = wmma_f32_16x16x4(a2[kt], bop, acc);
    }
    f2[nt] = acc;
  }
#pragma unroll
  for (int nt = 0; nt < 4; ++nt)
#pragma unroll
    for (int r = 0; r < 8; ++r)             // stage f[m][n] row-major
      st[(8 * khalf + r) * ROWP + nt * 16 + mrow] = f2[nt][r];

  // ---- x = e + f ; LayerNorm over H=64 (lane pair (l, l+16) per token) ----
  float x[32];
  float s1 = 0.f, s2 = 0.f;
  const int c0 = khalf * 32;
#pragma unroll
  for (int j = 0; j < 32; ++j) {
    const float xv = erow[c0 + j] + st[mrow * ROWP + c0 + j];
    x[j] = xv; s1 += xv; s2 += xv * xv;
  }
  s1 += __shfl_xor(s1, 16, 32);
  s2 += __shfl_xor(s2, 16, 32);
  const float mu  = s1 * (1.0f / HH);
  const float var = s2 * (1.0f / HH) - mu * mu;
  const float rs  = rsqrtf(var + 1e-5f);
  float* orow = hs + tok * HH + c0;
#pragma unroll
  for (int j = 0; j < 32; ++j)
    orow[j] = (x[j] - mu) * rs * lng[c0 + j] + lnb[c0 + j];
}

// ---------------------------------------------------------------------------
// Phase 2: sequential TTT scan, one wave32 per batch row (latency-bound).
// Lane l owns: w1 row l; w2 rows l & l+32 (row copy); w2 column l (col copy).
// LDS only used for 32-wide broadcasts (kh, hmid, d).
// kh/vh loads are software-pipelined one step ahead to hide L2 latency on the
// 2047-long serial dependence chain; the final prefetch (hs[b, L-1]) feeds the
// epilogue directly.
// ---------------------------------------------------------------------------
__global__ __launch_bounds__(32) void ttt_scan_kernel(
    const float* __restrict__ hs,
    const float* __restrict__ bw1, const float* __restrict__ bb1,
    const float* __restrict__ bw2, const float* __restrict__ bb2,
    const float* __restrict__ outw, const float* __restrict__ outb,
    float* __restrict__ out) {
  __shared__ float khlds[64];
  __shared__ float hmlds[32];
  __shared__ float dlds[64];
  const int b = blockIdx.x;
  const int l = threadIdx.x;

  float w1r[64];
#pragma unroll
  for (int j = 0; j < 64; ++j) w1r[j] = bw1[l * 64 + j];
  float w2a[32], w2b[32];
#pragma unroll
  for (int i = 0; i < 32; ++i) {
    w2a[i] = bw2[l * 32 + i];
    w2b[i] = bw2[(l + 32) * 32 + i];
  }
  float w2c[64];
#pragma unroll
  for (int r = 0; r < 64; ++r) w2c[r] = bw2[r * 32 + l];
  float b1l = bb1[l];
  float b2a = bb2[l], b2b = bb2[l + 32];

  const float* base = hs + (long)b * (LL * HH);

  // prologue: preload step-0 kh/vh
  float khA = base[l],          khB = base[l + 32];
  float vhA = base[HH + l],     vhB = base[HH + l + 32];

  for (int t = 0; t < TSTEPS; ++t) {
    khlds[l]      = khA;
    khlds[l + 32] = khB;
    const float cvA = vhA, cvB = vhB;

    // prefetch next step's kh/vh (t=TSTEPS-1 touches rows 4094/4095: in
    // bounds, and row 4095 is exactly hs[b, L-1] needed by the epilogue)
    {
      const float* kh2 = base + (2 * t + 2) * HH;
      const float* vh2 = base + (2 * t + 3) * HH;
      khA = kh2[l];  khB = kh2[l + 32];
      vhA = vh2[l];  vhB = vh2[l + 32];
    }
    __syncthreads();

    // z1 = w1 @ kh + b1   (lane l -> inner unit l)
    float z1 = b1l;
#pragma unroll
    for (int j = 0; j < 64; ++j) z1 = fmaf(w1r[j], khlds[j], z1);
    const float hm = fmaxf(z1, 0.0f);
    hmlds[l] = hm;
    __syncthreads();

    // pred = w2 @ hmid + b2  (lane l -> outputs l, l+32)
    float hmb[32];
    float pA = b2a, pB = b2b;
#pragma unroll
    for (int i = 0; i < 32; ++i) {
      const float h = hmlds[i];
      hmb[i] = h;
      pA = fmaf(w2a[i], h, pA);
      pB = fmaf(w2b[i], h, pB);
    }
    const float dA = (2.0f / HH) * (pA - cvA);
    const float dB = (2.0f / HH) * (pB - cvB);
    dlds[l]      = dA;
    dlds[l + 32] = dB;
    __syncthreads();

    // dh = (w2^T @ d) (pre-update w2), fused with column-copy update
    float dh = 0.f;
    const float chm = LRATE * hm;
#pragma unroll
    for (int r = 0; r < 64; ++r) {
      const float dr = dlds[r];
      dh = fmaf(w2c[r], dr, dh);
      w2c[r] = fmaf(-chm, dr, w2c[r]);       // w2[r][l] -= LR*d[r]*hm[l]
    }
    // row-copy update of w2
    const float cA = LRATE * dA, cB = LRATE * dB;
#pragma unroll
    for (int i = 0; i < 32; ++i) {
      w2a[i] = fmaf(-cA, hmb[i], w2a[i]);
      w2b[i] = fmaf(-cB, hmb[i], w2b[i]);
    }
    dh = (z1 > 0.0f) ? dh : 0.0f;
    const float cw = LRATE * dh;
#pragma unroll
    for (int j = 0; j < 64; ++j) w1r[j] = fmaf(-cw, khlds[j], w1r[j]);
    b1l -= LRATE * dh;
    b2a -= LRATE * dA;
    b2b -= LRATE * dB;
    __syncthreads();                         // protect khlds/dlds reuse
  }

  // ---- epilogue: ctx = w2 @ relu(w1 @ hs[b,-1] + b1) + b2 ; out = ctx@out_w^T+out_b
  // hs[b, L-1] is sitting in the last vh prefetch (vhA/vhB).
  khlds[l]      = vhA;
  khlds[l + 32] = vhB;
  __syncthreads();
  float z1 = b1l;
#pragma unroll
  for (int j = 0; j < 64; ++j) z1 = fmaf(w1r[j], khlds[j], z1);
  const float hm = fmaxf(z1, 0.0f);
  hmlds[l] = hm;
  __syncthreads();
  float cxA = b2a, cxB = b2b;
#pragma unroll
  for (int i = 0; i < 32; ++i) {
    const float h = hmlds[i];
    cxA = fmaf(w2a[i], h, cxA);
    cxB = fmaf(w2b[i], h, cxB);
  }
  dlds[l]      = cxA;                        // ctx broadcast
  dlds[l + 32] = cxB;
  __syncthreads();
  float oA = outb[l], oB = outb[l + 32];
  const float* ra = outw + l * HH;
  const float* rb = outw + (l + 32) * HH;
#pragma unroll
  for (int h = 0; h < 64; ++h) {
    const float c = dlds[h];
    oA = fmaf(ra[h], c, oA);
    oB = fmaf(rb[h], c, oB);
  }
  out[b * 64 + l]      = oA;
  out[b * 64 + l + 32] = oB;
}

extern "C" void kernel_launch(void* const* d_in, const int* in_sizes, int n_in,
                              void* d_out, int out_size, void* d_ws,
                              size_t ws_size, hipStream_t stream) {
  (void)in_sizes; (void)n_in; (void)out_size; (void)ws_size;
  const int*   seq   = (const int*)  d_in[0];
  const float* embed = (const float*)d_in[1];
  const float* ffw1  = (const float*)d_in[2];
  const float* ffb1  = (const float*)d_in[3];
  const float* ffw2  = (const float*)d_in[4];
  const float* ffb2  = (const float*)d_in[5];
  const float* lng   = (const float*)d_in[6];
  const float* lnb   = (const float*)d_in[7];
  const float* bw1   = (const float*)d_in[8];
  const float* bb1   = (const float*)d_in[9];
  const float* bw2   = (const float*)d_in[10];
  const float* bb2   = (const float*)d_in[11];
  const float* outw  = (const float*)d_in[12];
  const float* outb  = (const float*)d_in[13];
  float* hs  = (float*)d_ws;   // B*L*H f32 = 64 MiB staging
  float* out = (float*)d_out;

  // 262144 tokens / 16 per wave / 4 waves per block = 4096 blocks
  tok_hs_kernel<<<dim3((BBATCH * LL) / 16 / 4), dim3(128), 0, stream>>>(
      seq, embed, ffw1, ffb1, ffw2, ffb2, lng, lnb, hs);
  // one wave32 per batch row; the scan is the serial critical path
  ttt_scan_kernel<<<dim3(BBATCH), dim3(32), 0, stream>>>(
      hs, bw1, bb1, bw2, bb2, outw, outb, out);
}